// QuantizationLayerVoxGrid_824633721184
// MI455X (gfx1250) — compile-verified
//
#include <hip/hip_runtime.h>

// Voxel grid dims (from reference): C=10, H=480, W=640
#define C_DIM 10
#define H_DIM 480
#define W_DIM 640
#define HW    (W_DIM * H_DIM)         // 307200
#define CHW   (W_DIM * H_DIM * C_DIM) // 3072000

// ---------------------------------------------------------------------------
// K1: zero the 196.6 MB output with b128 stores (dominant memory cost),
//     and reset the tmax accumulator in workspace.
// ---------------------------------------------------------------------------
__global__ void vox_zero_kernel(float4* __restrict__ out4, int n4,
                                float* __restrict__ out_tail, int tail,
                                unsigned* __restrict__ ws) {
    int i = blockIdx.x * blockDim.x + threadIdx.x;
    const int stride = gridDim.x * blockDim.x;
    if (i == 0) ws[0] = 0u;  // tmax bits (t > 0 so uint-compare == float-compare)
    const float4 z = make_float4(0.f, 0.f, 0.f, 0.f);
    for (int j = i; j < n4; j += stride) out4[j] = z;      // global_store_b128
    for (int j = i; j < tail; j += stride) out_tail[j] = 0.f;
}

// ---------------------------------------------------------------------------
// K2: max over t = events[i*5 + 2].  wave32 shuffle reduce -> LDS (ds ops)
//     -> one global_atomic_max_u32 per block.  Dense stride-20B reads pull
//     full cachelines, so scalar t loads already saturate bandwidth.
// ---------------------------------------------------------------------------
__global__ void vox_tmax_kernel(const float* __restrict__ ev, int n,
                                unsigned* __restrict__ ws) {
    __shared__ float smax[8];  // 256 threads = 8 wave32s on gfx1250
    int i = blockIdx.x * blockDim.x + threadIdx.x;
    const int stride = gridDim.x * blockDim.x;
    float m = 0.f;
    for (int j = i; j < n; j += stride) m = fmaxf(m, ev[5 * j + 2]);

    // wave32 reduction (warpSize == 32 on CDNA5; do NOT hardcode 64)
    #pragma unroll
    for (int off = 16; off > 0; off >>= 1)
        m = fmaxf(m, __shfl_xor(m, off, 32));

    const int lane = threadIdx.x & 31;
    const int wave = threadIdx.x >> 5;
    if (lane == 0) smax[wave] = m;          // ds_store_b32
    __syncthreads();

    if (wave == 0) {
        m = (lane < (blockDim.x >> 5)) ? smax[lane] : 0.f;  // ds_load_b32
        #pragma unroll
        for (int off = 4; off > 0; off >>= 1)
            m = fmaxf(m, __shfl_xor(m, off, 32));
        if (lane == 0) atomicMax(ws, __float_as_uint(m));   // global_atomic_max_u32
    }
}

// ---------------------------------------------------------------------------
// K3 helper: bin + native FP32 scatter-add.
//   bin = ceil(t/tmax * C) - 1 ; valid iff 0 <= bin < C ; +1.0 at flat index.
//   (polarity is unused by the reference: it accumulates counts)
// ---------------------------------------------------------------------------
__device__ __forceinline__ void vox_accum(float x, float y, float t, float b,
                                          float tmax, float* __restrict__ out) {
    const float tn  = t / tmax;
    const int   bin = (int)ceilf(tn * (float)C_DIM) - 1;
    if (bin >= 0 && bin < C_DIM) {
        const int idx = (int)x + W_DIM * (int)y + HW * bin + CHW * (int)b;
        unsafeAtomicAdd(out + idx, 1.0f);   // global_atomic_add_f32 (no CAS loop)
    }
}

// ---------------------------------------------------------------------------
// K3: scatter.  4 records (80 B) per thread-iteration via 5 aligned
//     global_load_b128 (vs 20 scalar b32) to cut vmem issue pressure on the
//     atomic-carrying pass.  global_prefetch_b8 one group-stride ahead.
// ---------------------------------------------------------------------------
__global__ void vox_scatter_kernel(const float* __restrict__ ev, int n,
                                   const unsigned* __restrict__ ws,
                                   float* __restrict__ out) {
    const float tmax = __uint_as_float(ws[0]);
    int g = blockIdx.x * blockDim.x + threadIdx.x;
    const int gstride = gridDim.x * blockDim.x;
    const int ngroups = n >> 2;   // 4 records per group

    for (int j = g; j < ngroups; j += gstride) {
        const float4* r4 = (const float4*)(ev + 20 * j);  // 80k bytes -> 16B aligned
        // speculative prefetch of next group (OOB prefetch silently dropped)
        __builtin_prefetch(ev + 20 * (j + gstride), 0, 1);

        const float4 v0 = r4[0];  // x0 y0 t0 p0
        const float4 v1 = r4[1];  // b0 x1 y1 t1
        const float4 v2 = r4[2];  // p1 b1 x2 y2
        const float4 v3 = r4[3];  // t2 p2 b2 x3
        const float4 v4 = r4[4];  // y3 t3 p3 b3

        vox_accum(v0.x, v0.y, v0.z, v1.x, tmax, out);  // record 0
        vox_accum(v1.y, v1.z, v1.w, v2.y, tmax, out);  // record 1
        vox_accum(v2.z, v2.w, v3.x, v3.z, tmax, out);  // record 2
        vox_accum(v3.w, v4.x, v4.y, v4.w, tmax, out);  // record 3
    }

    // scalar tail (N % 4 records; empty for N = 2,000,000 but kept generic)
    for (int j = (ngroups << 2) + g; j < n; j += gstride) {
        const float* r = ev + 5 * j;
        vox_accum(r[0], r[1], r[2], r[4], tmax, out);
    }
}

// ---------------------------------------------------------------------------
// Host-side launcher.  All kernels on `stream`; serialized stream order gives
// zero -> tmax -> scatter dependencies.  Everything re-initialized per call.
// ---------------------------------------------------------------------------
extern "C" void kernel_launch(void* const* d_in, const int* in_sizes, int n_in,
                              void* d_out, int out_size, void* d_ws, size_t ws_size,
                              hipStream_t stream) {
    const float* ev = (const float*)d_in[0];   // events [N,5] float32
    const int n = in_sizes[0] / 5;             // N events
    float* out = (float*)d_out;                // B*C*H*W float32
    unsigned* ws = (unsigned*)d_ws;            // ws[0] = tmax bits

    const int n4   = out_size >> 2;            // float4 chunks
    const int tail = out_size & 3;
    float4* out4 = (float4*)d_out;
    float* out_tail = out + (n4 << 2);

    // K1: zero 196.6 MB output (the roofline-dominant pass) + reset ws
    vox_zero_kernel<<<4096, 256, 0, stream>>>(out4, n4, out_tail, tail, ws);

    // K2: tmax reduction over 2M t-values
    vox_tmax_kernel<<<2048, 256, 0, stream>>>(ev, n, ws);

    // K3: 2M scatter-adds (4 records / thread-iter, b128 event loads)
    vox_scatter_kernel<<<1024, 256, 0, stream>>>(ev, n, ws, out);
}